// MultiWaveletTransform_32126355374163
// MI455X (gfx1250) — compile-verified
//
#include <hip/hip_runtime.h>
#include <hip/hip_bf16.h>
#include <math.h>

// ---------------------------------------------------------------------------
// MultiWaveletTransform (FEDformer MWT_CZ) for MI455X / gfx1250.
// f32 end-to-end using V_WMMA_F32_16X16X4_F32 for all GEMM-like stages.
// The mode-mixing einsum is HBM-bound (2 flop/byte on fp32 weights) -> fp32
// WMMA + split-K across 8 segments (512 WGs) to saturate 23.3 TB/s, with
// GLOBAL_LOAD_ASYNC_TO_LDS_B128 staging (no VGPR round-trip, ASYNCcnt).
// ---------------------------------------------------------------------------

typedef float v2f __attribute__((ext_vector_type(2)));
typedef float v8f __attribute__((ext_vector_type(8)));

#define CKC 1024   // C_CH * K_WAV
#define NB  8      // batch
#define SEG 8      // split-K segments for the spectral einsum
#define LPAD 20    // einsum LDS row stride (floats): 16B-aligned quads + no bank conflicts

__device__ __forceinline__ v8f wmma_f32_k4(v2f a, v2f b, v8f c) {
  // D = A(16x4) * B(4x16) + C(16x16), all f32.
  return __builtin_amdgcn_wmma_f32_16x16x4_f32(false, a, false, b, (short)0, c,
                                               false, false);
}

// Async 16-byte global -> LDS copy (CDNA5 GLOBAL_LOAD_ASYNC_TO_LDS_B128).
// LDS byte address = low 32 bits of the generic pointer (ISA 10.2: LDS
// aperture address truncates to addr[31:0]).
__device__ __forceinline__ void async_g2l_b128(const float* gsrc,
                                               float* lds_dst) {
  unsigned int l = (unsigned int)(unsigned long long)lds_dst;
  asm volatile("global_load_async_to_lds_b128 %0, %1, off"
               :
               : "v"(l), "v"(gsrc)
               : "memory");
}
__device__ __forceinline__ void wait_async0() {
  asm volatile("s_wait_asynccnt 0x0" ::: "memory");
}

// ---------------------------------------------------------------------------
// Filter-bank generation (Legendre multiwavelets), double precision, 1 thread.
// Reproduces _legendre_filters(8) from the reference.
// out layout: EC_S[16*8] | EC_D[16*8] | RC_E[16*8] | RC_O[16*8]
// ---------------------------------------------------------------------------
__device__ static void d_compose(const double* p, int deg, double a, double b,
                                 double* o8) {
  double pw[8], nw[8];
  for (int j = 0; j < 8; ++j) { o8[j] = 0.0; pw[j] = 0.0; }
  pw[0] = 1.0;
  for (int i = 0; i <= deg; ++i) {
    for (int j = 0; j < 8; ++j) o8[j] += p[i] * pw[j];
    for (int j = 0; j < 8; ++j) nw[j] = a * pw[j];
    for (int j = 0; j < 7; ++j) nw[j + 1] += b * pw[j];
    for (int j = 0; j < 8; ++j) pw[j] = nw[j];
  }
}

__device__ static double d_inner(const double* a, int la, const double* b,
                                 int lb, bool half) {
  double prod[16];
  for (int i = 0; i < 16; ++i) prod[i] = 0.0;
  for (int i = 0; i < la; ++i)
    for (int j = 0; j < lb; ++j) prod[i + j] += a[i] * b[j];
  const int lp = la + lb - 1;
  double s = 0.0;
  for (int r = 0; r < lp; ++r) {
    double p = prod[r];
    if (fabs(p) < 1e-8) p = 0.0;
    double w = ldexp(1.0, -(r + 1));
    if (!half) w = 1.0 - w;
    s += p / (double)(r + 1) * w;
  }
  return s;
}

__device__ static double d_pev(const double* c, double x) {
  double r = 0.0;
  for (int i = 7; i >= 0; --i) r = r * x + c[i];
  return r;
}

__global__ void build_filters(float* __restrict__ out) {
  if (threadIdx.x != 0 || blockIdx.x != 0) return;

  // Legendre polynomials in power basis, deg <= 7
  double leg[8][8];
  for (int i = 0; i < 8; ++i)
    for (int j = 0; j < 8; ++j) leg[i][j] = 0.0;
  leg[0][0] = 1.0; leg[1][1] = 1.0;
  for (int n = 1; n < 7; ++n)
    for (int j = 0; j < 8; ++j) {
      double v = -(double)n * leg[n - 1][j];
      if (j > 0) v += (2.0 * n + 1.0) * leg[n][j - 1];
      leg[n + 1][j] = v / (double)(n + 1);
    }

  double phi[8][8], phi2[8][8];
  for (int ki = 0; ki < 8; ++ki) {
    double c1[8], c2[8];
    d_compose(leg[ki], ki, -1.0, 2.0, c1);   // P_ki(2x-1)
    d_compose(leg[ki], ki, -1.0, 4.0, c2);   // P_ki(4x-1)
    double s = sqrt(2.0 * ki + 1.0);
    for (int j = 0; j < 8; ++j) {
      phi[ki][j]  = s * c1[j];
      phi2[ki][j] = sqrt(2.0) * s * c2[j];
    }
  }

  double psi1[8][8], psi2[8][8];
  for (int i = 0; i < 8; ++i)
    for (int j = 0; j < 8; ++j) { psi1[i][j] = 0.0; psi2[i][j] = 0.0; }
  for (int ki = 0; ki < 8; ++ki) {
    for (int j = 0; j < 8; ++j) psi1[ki][j] = phi2[ki][j];
    for (int i = 0; i < 8; ++i) {
      double p = d_inner(phi2[ki], ki + 1, phi[i], i + 1, true);
      for (int j = 0; j < 8; ++j) {
        psi1[ki][j] -= p * phi[i][j];
        psi2[ki][j] -= p * phi[i][j];
      }
    }
    for (int jj = 0; jj < ki; ++jj) {
      double p = d_inner(phi2[ki], ki + 1, psi1[jj], 8, true);
      for (int j = 0; j < 8; ++j) {
        psi1[ki][j] -= p * psi1[jj][j];
        psi2[ki][j] -= p * psi2[jj][j];
      }
    }
    double n1 = d_inner(psi1[ki], 8, psi1[ki], 8, true);
    double n2 = d_inner(psi2[ki], 8, psi2[ki], 8, false);
    double nrm = sqrt(n1 + n2);
    for (int j = 0; j < 8; ++j) {
      psi1[ki][j] /= nrm;  psi2[ki][j] /= nrm;
      if (fabs(psi1[ki][j]) < 1e-8) psi1[ki][j] = 0.0;
      if (fabs(psi2[ki][j]) < 1e-8) psi2[ki][j] = 0.0;
    }
  }

  // 8-point Gauss-Legendre on [-1,1]
  const double gx[8] = {-0.96028985649753623, -0.79666647741362674,
                        -0.52553240991632899, -0.18343464249564980,
                         0.18343464249564980,  0.52553240991632899,
                         0.79666647741362674,  0.96028985649753623};
  const double gw[8] = { 0.10122853629037626,  0.22238103445337447,
                         0.31370664587788729,  0.36268378337836198,
                         0.36268378337836198,  0.31370664587788729,
                         0.22238103445337447,  0.10122853629037626};
  const double s2 = 1.0 / sqrt(2.0);
  double H0[8][8], H1[8][8], G0[8][8], G1[8][8];
  for (int ki = 0; ki < 8; ++ki)
    for (int kp = 0; kp < 8; ++kp) {
      double h0 = 0, h1 = 0, g0 = 0, g1 = 0;
      for (int q = 0; q < 8; ++q) {
        double xm = (gx[q] + 1.0) * 0.5;
        double wm = gw[q] * 0.5;
        double pk = d_pev(phi[kp], xm);
        h0 += wm * d_pev(phi[ki],  xm * 0.5) * pk;         // x/2 in (0,.5)
        g0 += wm * d_pev(psi1[ki], xm * 0.5) * pk;         // psi1 branch
        h1 += wm * d_pev(phi[ki],  (xm + 1.0) * 0.5) * pk; // (x+1)/2 in (.5,1)
        g1 += wm * d_pev(psi2[ki], (xm + 1.0) * 0.5) * pk; // psi2 branch
      }
      h0 *= s2; h1 *= s2; g0 *= s2; g1 *= s2;
      H0[ki][kp] = (fabs(h0) < 1e-8) ? 0.0 : h0;
      H1[ki][kp] = (fabs(h1) < 1e-8) ? 0.0 : h1;
      G0[ki][kp] = (fabs(g0) < 1e-8) ? 0.0 : g0;
      G1[ki][kp] = (fabs(g1) < 1e-8) ? 0.0 : g1;
    }

  float* EC_S = out;        // concat(H0.T, H1.T)  [16][8]
  float* EC_D = out + 128;  // concat(G0.T, G1.T)
  float* RC_E = out + 256;  // concat(H0, G0)
  float* RC_O = out + 384;  // concat(H1, G1)
  for (int kk = 0; kk < 8; ++kk)
    for (int j = 0; j < 8; ++j) {
      EC_S[j * 8 + kk]       = (float)H0[kk][j];
      EC_S[(8 + j) * 8 + kk] = (float)H1[kk][j];
      EC_D[j * 8 + kk]       = (float)G0[kk][j];
      EC_D[(8 + j) * 8 + kk] = (float)G1[kk][j];
      RC_E[j * 8 + kk]       = (float)H0[j][kk];
      RC_E[(8 + j) * 8 + kk] = (float)G0[j][kk];
      RC_O[j * 8 + kk]       = (float)H1[j][kk];
      RC_O[(8 + j) * 8 + kk] = (float)G1[j][kk];
    }
}

// ---------------------------------------------------------------------------
// WMMA f32 GEMM:  C[row, o] = sum_k A[row, k] * W[o, k] + bias[o]
// Block: 128 threads = 4 waves. Tile M=16, N=64 (16 per wave), K-chunk 32.
// Tiles are staged with async global->LDS b128 (row stride 36 floats = 144B
// keeps every quad 16B-aligned; fragment reads stay bank-conflict-free).
// ---------------------------------------------------------------------------
__global__ __launch_bounds__(128)
void wmma_gemm_f32(const float* __restrict__ A, const float* __restrict__ W,
                   const float* __restrict__ bias, float* __restrict__ C,
                   int N, int K) {
  __shared__ float As[16][36];
  __shared__ float Ws[64][36];
  const int tid  = threadIdx.x;
  const int wave = tid >> 5;
  const int lane = tid & 31;
  const int m0   = blockIdx.x * 16;
  const int n0   = blockIdx.y * 64;
  const int row  = lane & 15;
  const int hi   = lane >> 4;

  v8f acc = {};
  for (int k0 = 0; k0 < K; k0 += 32) {
    {  // A tile: 16x32 = 128 quads, exactly one per thread
      int r = tid >> 3, cq = (tid & 7) * 4;
      async_g2l_b128(&A[(size_t)(m0 + r) * K + (k0 + cq)], &As[r][cq]);
    }
#pragma unroll
    for (int it = 0; it < 4; ++it) {  // W tile: 64x32 = 512 quads, 4/thread
      int idx = tid + it * 128;
      int r = idx >> 3, cq = (idx & 7) * 4;
      async_g2l_b128(&W[(size_t)(n0 + r) * K + (k0 + cq)], &Ws[r][cq]);
    }
    wait_async0();
    __syncthreads();
#pragma unroll
    for (int kk = 0; kk < 32; kk += 4) {
      v2f a, b;
      a.x = As[row][kk + 2 * hi + 0];
      a.y = As[row][kk + 2 * hi + 1];
      b.x = Ws[wave * 16 + row][kk + 2 * hi + 0];
      b.y = Ws[wave * 16 + row][kk + 2 * hi + 1];
      acc = wmma_f32_k4(a, b, acc);
    }
    __syncthreads();
  }
  const int ocol = n0 + wave * 16 + row;
  const float bv = bias[ocol];
#pragma unroll
  for (int r = 0; r < 8; ++r) {
    int orow = m0 + r + 8 * hi;
    C[(size_t)orow * N + ocol] = acc[r] + bv;
  }
}

// ---------------------------------------------------------------------------
// Decomposition: s = xa @ EC_S, d = xa @ EC_D where xa = [x_even | x_odd].
// ---------------------------------------------------------------------------
__global__ void decompose_kernel(const float* __restrict__ x,
                                 const float* __restrict__ EC_S,
                                 const float* __restrict__ EC_D,
                                 float* __restrict__ s, float* __restrict__ d,
                                 int N2) {
  size_t t = (size_t)blockIdx.x * blockDim.x + threadIdx.x;
  int kk = (int)(t & 7);
  int c  = (int)((t >> 3) & 127);
  size_t n2 = (t >> 10) % (size_t)N2;
  size_t b  = t / ((size_t)N2 * CKC);
  const float* xe = x + ((b * (size_t)(2 * N2)) + 2 * n2) * CKC + c * 8;
  const float* xo = xe + CKC;
  float sv = 0.f, dv = 0.f;
#pragma unroll
  for (int j = 0; j < 8; ++j) {
    float ve = xe[j], vo = xo[j];
    sv += ve * EC_S[j * 8 + kk] + vo * EC_S[(8 + j) * 8 + kk];
    dv += ve * EC_D[j * 8 + kk] + vo * EC_D[(8 + j) * 8 + kk];
  }
  s[t] = sv;
  d[t] = dv;
}

// ---------------------------------------------------------------------------
// Truncated forward DFT: X[b,m,i] = sum_n y[b,n,i] * exp(-2*pi*i*m*n/N)
// ---------------------------------------------------------------------------
__global__ void dft_fwd(const float* __restrict__ y, float* __restrict__ Xre,
                        float* __restrict__ Xim, int N, int ln) {
  int i = blockIdx.x * blockDim.x + threadIdx.x;  // channel
  int m = blockIdx.y;
  int b = blockIdx.z;
  const float* yp = y + (size_t)b * N * CKC + i;
  float w = -2.0f * 3.14159265358979323846f * (float)m / (float)N;
  float re = 0.f, im = 0.f;
  for (int n = 0; n < N; ++n) {
    float sn, cs;
    __sincosf(w * (float)n, &sn, &cs);
    float v = yp[(size_t)n * CKC];
    re += v * cs;
    im += v * sn;
  }
  size_t o = ((size_t)b * ln + m) * CKC + i;
  Xre[o] = re;
  Xim[o] = im;
}

// ---------------------------------------------------------------------------
// Mode-mixing einsum with f32 WMMA (the HBM-bound hot loop), split-K.
//   Ud[b,o,m] = sum_i Xd[b,i,m]*wA[i,o,m] + Xs[b,i,m]*wB[i,o,m]   (complex)
//   Us[b,o,m] = sum_i Xd[b,i,m]*wC[i,o,m]                          (complex)
// Complex via stacked-A trick: A(16xK) rows 0..7 = Re (b), rows 8..15 = Im.
//   P = A@Wre, Q = A@Wim; out_re = P[0:8]-Q[8:16]; out_im = Q[0:8]+P[8:16].
// One WG = 16 waves = all 16 modes for one 16-wide o-tile; grid.y = SEG
// split-K segments (512 WGs) so HBM stays saturated. Weight staging: one
// async b128 global->LDS copy per thread per array per chunk, drained with a
// single s_wait_asynccnt; LDS row stride LPAD=20 floats keeps quads
// 16B-aligned and fragment reads bank-conflict-free.
// ---------------------------------------------------------------------------
__global__ __launch_bounds__(512)
void spec_einsum(const float* __restrict__ Xd_re, const float* __restrict__ Xd_im,
                 const float* __restrict__ Xs_re, const float* __restrict__ Xs_im,
                 const float* __restrict__ wA_re, const float* __restrict__ wA_im,
                 const float* __restrict__ wB_re, const float* __restrict__ wB_im,
                 const float* __restrict__ wC_re, const float* __restrict__ wC_im,
                 float* __restrict__ Pd_re, float* __restrict__ Pd_im,
                 float* __restrict__ Ps_re, float* __restrict__ Ps_im,
                 int ln) {
  __shared__ float LA_re[8 * 16 * LPAD], LA_im[8 * 16 * LPAD];
  __shared__ float LB_re[8 * 16 * LPAD], LB_im[8 * 16 * LPAD];
  __shared__ float LC_re[8 * 16 * LPAD], LC_im[8 * 16 * LPAD];
  const int tid  = threadIdx.x;
  const int wave = tid >> 5;          // = mode m
  const int lane = tid & 31;
  const int o0   = blockIdx.x * 16;
  const int seg  = blockIdx.y;        // split-K segment
  const int row  = lane & 15;
  const int hi   = lane >> 4;
  const int bidx = row & 7;
  const bool isIm  = (row >= 8);
  const int  m     = wave;
  const bool active = (m < ln);

  const int iBeg = seg * (CKC / SEG);
  const int iEnd = iBeg + (CKC / SEG);

  // Per-thread staging coordinates: one b128 per array per 8-i chunk.
  const int sII = tid >> 6;            // 0..7   (i within chunk)
  const int sOL = (tid & 63) >> 2;     // 0..15  (o within tile)
  const int sMQ = (tid & 3) * 4;       // 0,4,8,12 (mode quad)
  const int sL  = (sII * 16 + sOL) * LPAD + sMQ;

  v8f PA = {}, QA = {}, PC = {}, QC = {};

  for (int i0 = iBeg; i0 < iEnd; i0 += 8) {
    {
      size_t g = ((size_t)(i0 + sII) * CKC + (o0 + sOL)) * 16 + sMQ;
      async_g2l_b128(wA_re + g, &LA_re[sL]);
      async_g2l_b128(wA_im + g, &LA_im[sL]);
      async_g2l_b128(wB_re + g, &LB_re[sL]);
      async_g2l_b128(wB_im + g, &LB_im[sL]);
      async_g2l_b128(wC_re + g, &LC_re[sL]);
      async_g2l_b128(wC_im + g, &LC_im[sL]);
    }
    wait_async0();
    __syncthreads();
    if (active) {
      const float* Xd = isIm ? Xd_im : Xd_re;
      const float* Xs = isIm ? Xs_im : Xs_re;
      const size_t xoff = ((size_t)bidx * ln + m) * CKC;
#pragma unroll
      for (int kk = 0; kk < 8; kk += 4) {
        const int ia = i0 + kk + 2 * hi;
        v2f ad, as;
        ad.x = Xd[xoff + ia];  ad.y = Xd[xoff + ia + 1];
        as.x = Xs[xoff + ia];  as.y = Xs[xoff + ia + 1];
        const int k0l = kk + 2 * hi;
        const int lb0 = (k0l * 16 + row) * LPAD + m;
        const int lb1 = ((k0l + 1) * 16 + row) * LPAD + m;
        v2f bre, bim;
        bre.x = LA_re[lb0];  bre.y = LA_re[lb1];
        bim.x = LA_im[lb0];  bim.y = LA_im[lb1];
        PA = wmma_f32_k4(ad, bre, PA);
        QA = wmma_f32_k4(ad, bim, QA);
        bre.x = LB_re[lb0];  bre.y = LB_re[lb1];
        bim.x = LB_im[lb0];  bim.y = LB_im[lb1];
        PA = wmma_f32_k4(as, bre, PA);
        QA = wmma_f32_k4(as, bim, QA);
        bre.x = LC_re[lb0];  bre.y = LC_re[lb1];
        bim.x = LC_im[lb0];  bim.y = LC_im[lb1];
        PC = wmma_f32_k4(ad, bre, PC);
        QC = wmma_f32_k4(ad, bim, QC);
      }
    }
    __syncthreads();
  }

  if (active) {
    const size_t segStride = (size_t)NB * ln * CKC;
#pragma unroll
    for (int r = 0; r < 8; ++r) {
      float pa = PA[r], qa = QA[r], pc = PC[r], qc = QC[r];
      float paw = __shfl_xor(pa, 16);
      float qaw = __shfl_xor(qa, 16);
      float pcw = __shfl_xor(pc, 16);
      float qcw = __shfl_xor(qc, 16);
      if (lane < 16) {   // lanes 0..15 hold rows 0..7 (the Re-part rows), b = r
        size_t o = (size_t)seg * segStride + ((size_t)r * ln + m) * CKC + o0 + lane;
        Pd_re[o] = pa - qaw;
        Pd_im[o] = qa + paw;
        Ps_re[o] = pc - qcw;
        Ps_im[o] = qc + pcw;
      }
    }
  }
}

// Deterministic fixed-order reduction of split-K partials (4 arrays at once).
__global__ void reduce_partials(const float* __restrict__ P0,
                                const float* __restrict__ P1,
                                const float* __restrict__ P2,
                                const float* __restrict__ P3,
                                float* __restrict__ O0, float* __restrict__ O1,
                                float* __restrict__ O2, float* __restrict__ O3,
                                size_t n) {
  size_t t = (size_t)blockIdx.x * blockDim.x + threadIdx.x;
  if (t >= n) return;
  float s0 = 0.f, s1 = 0.f, s2 = 0.f, s3 = 0.f;
#pragma unroll
  for (int s = 0; s < SEG; ++s) {
    size_t o = (size_t)s * n + t;
    s0 += P0[o];  s1 += P1[o];  s2 += P2[o];  s3 += P3[o];
  }
  O0[t] = s0;  O1[t] = s1;  O2[t] = s2;  O3[t] = s3;
}

// ---------------------------------------------------------------------------
// Truncated inverse rfft (only ln nonzero modes):
//   u[b,n,o] = (1/N) * sum_m c_m * (Ure*cos(2pi m n/N) - Uim*sin(...))
// ---------------------------------------------------------------------------
__global__ void dft_inv(const float* __restrict__ Ure,
                        const float* __restrict__ Uim,
                        float* __restrict__ u, int N, int ln) {
  int i = blockIdx.x * blockDim.x + threadIdx.x;
  int n = blockIdx.y;
  int b = blockIdx.z;
  float acc = 0.f;
  for (int m = 0; m < ln; ++m) {
    size_t o = ((size_t)b * ln + m) * CKC + i;
    float re = Ure[o], im = Uim[o];
    if (2 * m == N) {
      acc += ((n & 1) ? -re : re);          // Nyquist: real part only
    } else {
      float sn, cs;
      __sincosf(2.0f * 3.14159265358979323846f * (float)m * (float)n / (float)N,
                &sn, &cs);
      float cm = (m == 0) ? 1.f : 2.f;
      acc += cm * (re * cs - im * sn);
    }
  }
  u[((size_t)b * N + n) * CKC + i] = acc / (float)N;
}

// ---------------------------------------------------------------------------
// Coarsest-scale transform: y = x @ T0_w.T + T0_b  on (B, 1, 128, 8)
// ---------------------------------------------------------------------------
__global__ void t0_apply(const float* __restrict__ x,
                         const float* __restrict__ T0_w,
                         const float* __restrict__ T0_b,
                         float* __restrict__ y) {
  int t = blockIdx.x * blockDim.x + threadIdx.x;
  if (t >= NB * CKC) return;
  int kk = t & 7;
  int c  = (t >> 3) & 127;
  int b  = t >> 10;
  const float* xp = x + (size_t)b * CKC + c * 8;
  float acc = T0_b[kk];
#pragma unroll
  for (int j = 0; j < 8; ++j) acc += xp[j] * T0_w[kk * 8 + j];
  y[t] = acc;
}

// ---------------------------------------------------------------------------
// Reconstruction: xp = x + Us; xc = [xp | Ud];
//   x_out[2r]   = xc @ RC_E ; x_out[2r+1] = xc @ RC_O
// ---------------------------------------------------------------------------
__global__ void recon_kernel(const float* __restrict__ xin,
                             const float* __restrict__ Us,
                             const float* __restrict__ Ud,
                             const float* __restrict__ RC_E,
                             const float* __restrict__ RC_O,
                             float* __restrict__ xout, int n) {
  size_t t = (size_t)blockIdx.x * blockDim.x + threadIdx.x;
  int kk = (int)(t & 7);
  int c  = (int)((t >> 3) & 127);
  size_t nr = (t >> 10) % (size_t)n;
  size_t b  = t / ((size_t)n * CKC);
  size_t base = (b * (size_t)n + nr) * CKC + c * 8;
  float e = 0.f, o = 0.f;
#pragma unroll
  for (int j = 0; j < 8; ++j) {
    float xc = xin[base + j] + Us[base + j];
    e += xc * RC_E[j * 8 + kk];
    o += xc * RC_O[j * 8 + kk];
  }
#pragma unroll
  for (int j = 0; j < 8; ++j) {
    float dv = Ud[base + j];
    e += dv * RC_E[(8 + j) * 8 + kk];
    o += dv * RC_O[(8 + j) * 8 + kk];
  }
  size_t ob = (b * (size_t)(2 * n) + 2 * nr) * CKC + c * 8 + kk;
  xout[ob]       = e;
  xout[ob + CKC] = o;
}

// ---------------------------------------------------------------------------
// Host orchestration
// ---------------------------------------------------------------------------
extern "C" void kernel_launch(void* const* d_in, const int* in_sizes, int n_in,
                              void* d_out, int out_size, void* d_ws,
                              size_t ws_size, hipStream_t stream) {
  (void)in_sizes; (void)n_in; (void)out_size; (void)ws_size;
  const float* values = (const float*)d_in[2];   // queries/keys unused by ref
  const float* Lk0_w  = (const float*)d_in[3];
  const float* Lk0_b  = (const float*)d_in[4];
  const float* Lk1_w  = (const float*)d_in[5];
  const float* Lk1_b  = (const float*)d_in[6];
  const float* T0_w   = (const float*)d_in[7];
  const float* T0_b   = (const float*)d_in[8];
  const float* wA_re  = (const float*)d_in[9];
  const float* wA_im  = (const float*)d_in[10];
  const float* wB_re  = (const float*)d_in[11];
  const float* wB_im  = (const float*)d_in[12];
  const float* wC_re  = (const float*)d_in[13];
  const float* wC_im  = (const float*)d_in[14];
  float* ws  = (float*)d_ws;
  float* out = (float*)d_out;

  // ---- workspace layout (floats), ~190 MB total ----
  size_t off = 0;
  auto take = [&](size_t nfl) {
    size_t o = off;
    off += (nfl + 63) & ~(size_t)63;
    return o;
  };
  const size_t oF   = take(512);                     // filter banks
  const size_t oX0  = take((size_t)NB * 1024 * CKC); // Lk0 out / recon ping
  const size_t PYR  = (size_t)NB * 1023 * CKC;
  const size_t oS   = take(PYR);                     // s pyramid
  const size_t oD   = take(PYR);                     // d pyramid (-> Ud in place)
  const size_t oU   = take(PYR);                     // Us pyramid
  const size_t FR   = (size_t)NB * 16 * CKC;
  const size_t oXdr = take(FR), oXdi = take(FR);
  const size_t oXsr = take(FR), oXsi = take(FR);
  const size_t oUdr = take(FR), oUdi = take(FR);
  const size_t oUsr = take(FR), oUsi = take(FR);
  const size_t oPdr = take(FR * SEG), oPdi = take(FR * SEG);  // split-K partials
  const size_t oPsr = take(FR * SEG), oPsi = take(FR * SEG);
  const size_t oXT  = take((size_t)NB * CKC);        // coarsest x after T0
  const size_t oR1  = take((size_t)NB * 1024 * CKC); // recon pong

  float* EC_S = ws + oF;
  float* EC_D = EC_S + 128;
  float* RC_E = EC_S + 256;
  float* RC_O = EC_S + 384;

  build_filters<<<1, 1, 0, stream>>>(ws + oF);

  // Lk0: (8192 x 512) @ (1024 x 512)^T + b -> (8192 x 1024)
  wmma_gemm_f32<<<dim3(8192 / 16, 1024 / 64), 128, 0, stream>>>(
      values, Lk0_w, Lk0_b, ws + oX0, 1024, 512);

  size_t pOff[10];
  {
    size_t a = 0;
    for (int t = 0; t < 10; ++t) {
      pOff[t] = a;
      a += (size_t)NB * (1024 >> (t + 1)) * CKC;
    }
  }

  // ---- decomposition: 10 levels ----
  for (int t = 0; t < 10; ++t) {
    const int N2 = 1024 >> (t + 1);
    const int Kf = N2 / 2 + 1;
    const int ln = (Kf < 16) ? Kf : 16;
    const float* xin = (t == 0) ? (ws + oX0) : (ws + oS + pOff[t - 1]);
    float* sP = ws + oS + pOff[t];
    float* dP = ws + oD + pOff[t];
    float* uP = ws + oU + pOff[t];
    size_t tot = (size_t)NB * N2 * CKC;
    decompose_kernel<<<(unsigned)(tot / 256), 256, 0, stream>>>(
        xin, EC_S, EC_D, sP, dP, N2);
    dft_fwd<<<dim3(CKC / 256, ln, NB), 256, 0, stream>>>(dP, ws + oXdr,
                                                         ws + oXdi, N2, ln);
    dft_fwd<<<dim3(CKC / 256, ln, NB), 256, 0, stream>>>(sP, ws + oXsr,
                                                         ws + oXsi, N2, ln);
    // split-K spectral einsum: 64 o-tiles x SEG segments = 512 WGs
    spec_einsum<<<dim3(CKC / 16, SEG), 512, 0, stream>>>(
        ws + oXdr, ws + oXdi, ws + oXsr, ws + oXsi, wA_re, wA_im, wB_re, wB_im,
        wC_re, wC_im, ws + oPdr, ws + oPdi, ws + oPsr, ws + oPsi, ln);
    const size_t nred = (size_t)NB * ln * CKC;
    reduce_partials<<<(unsigned)((nred + 255) / 256), 256, 0, stream>>>(
        ws + oPdr, ws + oPdi, ws + oPsr, ws + oPsi, ws + oUdr, ws + oUdi,
        ws + oUsr, ws + oUsi, nred);
    // d slot becomes Ud; Us slot fills the Us pyramid
    dft_inv<<<dim3(CKC / 256, N2, NB), 256, 0, stream>>>(ws + oUdr, ws + oUdi,
                                                         dP, N2, ln);
    dft_inv<<<dim3(CKC / 256, N2, NB), 256, 0, stream>>>(ws + oUsr, ws + oUsi,
                                                         uP, N2, ln);
  }

  // coarsest-scale linear map
  t0_apply<<<32, 256, 0, stream>>>(ws + oS + pOff[9], T0_w, T0_b, ws + oXT);

  // ---- reconstruction: 10 levels, ping-pong buffers ----
  const float* cur = ws + oXT;
  float* ping[2] = {ws + oR1, ws + oX0};
  int pb = 0;
  for (int i = 9; i >= 0; --i) {
    int n = 1024 >> (i + 1);
    float* xo = ping[pb];
    pb ^= 1;
    size_t tot = (size_t)NB * n * CKC;
    recon_kernel<<<(unsigned)(tot / 256), 256, 0, stream>>>(
        cur, ws + oU + pOff[i], ws + oD + pOff[i], RC_E, RC_O, xo, n);
    cur = xo;
  }
  // 10 steps starting at ping[0] -> final full-length x lives in ws + oX0

  // Lk1: (8192 x 1024) @ (512 x 1024)^T + b -> d_out (B, L, 8, 64)
  wmma_gemm_f32<<<dim3(8192 / 16, 512 / 64), 128, 0, stream>>>(
      cur, Lk1_w, Lk1_b, out, 512, 1024);
}